// Classifier_Transformer_17016660427427
// MI455X (gfx1250) — compile-verified
//
#include <hip/hip_runtime.h>

typedef __attribute__((ext_vector_type(2))) float v2f;
typedef __attribute__((ext_vector_type(8))) float v8f;

constexpr int kN     = 20000;
constexpr int kE     = 320000;
constexpr int kDim   = 32;
constexpr int kHeads = 3;
constexpr int kHC    = 96;       // kHeads * kDim
constexpr int kG     = 16;
constexpr int kTiles = kN / 16;  // 1250 exact

// LDS weight staging layout (floats)
constexpr int kWqOff = 0;                    // 32*96
constexpr int kWkOff = kWqOff + kDim * kHC;  // 3072
constexpr int kWvOff = kWkOff + kDim * kHC;  // 6144
constexpr int kWsOff = kWvOff + kDim * kHC;  // 9216
constexpr int kBqOff = kWsOff + kDim * kDim; // 10240
constexpr int kBkOff = kBqOff + kHC;
constexpr int kBvOff = kBkOff + kHC;
constexpr int kBsOff = kBvOff + kHC;
constexpr int kLdsFloats = kBsOff + kDim;    // 10560 floats = 42.24 KB

__device__ __forceinline__ void stage(float* __restrict__ dst,
                                      const float* __restrict__ src,
                                      int nelem, int tid, int nthr) {
  for (int i = tid; i < nelem; i += nthr) dst[i] = src[i];
}

// ---------- WMMA tile GEMM: [16 x 32] x [32 x W] + bias -> [16 x W] ----------
// A (32-bit 16x4, ISA 7.12.2): lane&15 = row M; lane>>4 picks K pair {0,1}/{2,3}.
// B (4x16) mirrored: lane&15 = column N. C/D: VGPR r -> (M = r + 8*half, N).
template <int W>
__device__ __forceinline__ void tile_gemm(const v2f (&afrag)[8],
                                          const float* Wmat,   // LDS
                                          const float* bias,   // LDS
                                          float* __restrict__ dst,
                                          int row0, int lane) {
  const int col16 = lane & 15;
  const int half  = lane >> 4;
#pragma unroll
  for (int c = 0; c < W / 16; ++c) {
    const int col = c * 16 + col16;
    const float bv = bias[col];
    v8f acc;
#pragma unroll
    for (int r = 0; r < 8; ++r) acc[r] = bv;  // bias is per-column
#pragma unroll
    for (int s = 0; s < 8; ++s) {             // K = 32 in steps of 4
      const int kb = 4 * s + 2 * half;
      v2f b;
      b[0] = Wmat[kb * W + col];
      b[1] = Wmat[(kb + 1) * W + col];
      acc = __builtin_amdgcn_wmma_f32_16x16x4_f32(false, afrag[s], false, b,
                                                  (short)0, acc, false, false);
    }
#pragma unroll
    for (int r = 0; r < 8; ++r)
      dst[(size_t)(row0 + r + 8 * half) * W + col] = acc[r];
  }
}

// ---------- K1: projections q,k,v (N x 96) and s = x@Ws + bs (N x 32) ----------
__global__ void __launch_bounds__(128)
proj_kernel(const float* __restrict__ x,
            const float* __restrict__ Wq, const float* __restrict__ bq,
            const float* __restrict__ Wk, const float* __restrict__ bk,
            const float* __restrict__ Wv, const float* __restrict__ bv,
            const float* __restrict__ Ws, const float* __restrict__ bs,
            float* __restrict__ q, float* __restrict__ k,
            float* __restrict__ v, float* __restrict__ s) {
  __shared__ float smem[kLdsFloats];
  const int tid  = (int)threadIdx.x;
  const int nthr = (int)blockDim.x;
  stage(smem + kWqOff, Wq, kDim * kHC, tid, nthr);
  stage(smem + kWkOff, Wk, kDim * kHC, tid, nthr);
  stage(smem + kWvOff, Wv, kDim * kHC, tid, nthr);
  stage(smem + kWsOff, Ws, kDim * kDim, tid, nthr);
  stage(smem + kBqOff, bq, kHC, tid, nthr);
  stage(smem + kBkOff, bk, kHC, tid, nthr);
  stage(smem + kBvOff, bv, kHC, tid, nthr);
  stage(smem + kBsOff, bs, kDim, tid, nthr);
  __syncthreads();

  const int wave = (int)((blockIdx.x * blockDim.x + threadIdx.x) >> 5);
  const int lane = tid & 31;
  if (wave >= kTiles) return;  // wave-uniform; EXEC stays all-ones for WMMA
  const int row0 = wave * 16;
  const int row  = row0 + (lane & 15);
  const int half = lane >> 4;
  v2f a[8];
#pragma unroll
  for (int t = 0; t < 8; ++t) {
    a[t][0] = x[(size_t)row * kDim + 4 * t + 2 * half + 0];
    a[t][1] = x[(size_t)row * kDim + 4 * t + 2 * half + 1];
  }
  tile_gemm<kHC>(a, smem + kWqOff, smem + kBqOff, q, row0, lane);
  tile_gemm<kHC>(a, smem + kWkOff, smem + kBkOff, k, row0, lane);
  tile_gemm<kHC>(a, smem + kWvOff, smem + kBvOff, v, row0, lane);
  tile_gemm<kDim>(a, smem + kWsOff, smem + kBsOff, s, row0, lane);
}

// ---------- CSR build over incoming edges (dst is launch-invariant) ----------
__global__ void deg_kernel(const int* __restrict__ ei, int* __restrict__ deg) {
  const int e = (int)(blockIdx.x * blockDim.x + threadIdx.x);
  if (e >= kE) return;
  atomicAdd(&deg[ei[kE + e]], 1);
}

// single-block exclusive scan of deg[0..kN) -> rowptr[0..kN]
__global__ void __launch_bounds__(1024)
scan_kernel(const int* __restrict__ deg, int* __restrict__ rowptr) {
  constexpr int CH = (kN + 1023) / 1024;  // 20
  __shared__ int part[1024];
  const int t = (int)threadIdx.x;
  const int base = t * CH;
  int s = 0;
#pragma unroll
  for (int i = 0; i < CH; ++i)
    if (base + i < kN) s += deg[base + i];
  part[t] = s;
  __syncthreads();
  for (int off = 1; off < 1024; off <<= 1) {  // Hillis-Steele inclusive scan
    const int val = (t >= off) ? part[t - off] : 0;
    __syncthreads();
    part[t] += val;
    __syncthreads();
  }
  int run = (t > 0) ? part[t - 1] : 0;  // exclusive prefix
#pragma unroll
  for (int i = 0; i < CH; ++i) {
    if (base + i < kN) {
      rowptr[base + i] = run;
      run += deg[base + i];
    }
  }
  if (t == 0) rowptr[kN] = part[1023];
}

__global__ void fill_kernel(const int* __restrict__ ei, int* __restrict__ cursor,
                            int* __restrict__ esrc) {
  const int e = (int)(blockIdx.x * blockDim.x + threadIdx.x);
  if (e >= kE) return;
  const int dst = ei[kE + e];
  const int pos = atomicAdd(&cursor[dst], 1);
  esrc[pos] = ei[e];
}

// ---------- fused attention: one wave per node, lanes = 32 dims ----------
// online softmax per head over the node's incoming edges; no atomics.
__global__ void __launch_bounds__(256)
attn_kernel(const int* __restrict__ rowptr, const int* __restrict__ esrc,
            const float* __restrict__ q, const float* __restrict__ k,
            const float* __restrict__ v, const float* __restrict__ s,
            float* __restrict__ x, float* __restrict__ mid, int emit_mid) {
  const int n    = (int)((blockIdx.x * blockDim.x + threadIdx.x) >> 5);
  const int lane = (int)(threadIdx.x & 31);
  if (n >= kN) return;
  const int beg = rowptr[n];
  const int end = rowptr[n + 1];
  float acc = 0.0f;  // sum over heads of normalized output, dim = lane
#pragma unroll
  for (int h = 0; h < kHeads; ++h) {
    const float qd = q[(size_t)n * kHC + h * kDim + lane];
    float m = -INFINITY, z = 0.0f, o = 0.0f;
    for (int j = beg; j < end; ++j) {
      const int src = esrc[j];
      float p = qd * k[(size_t)src * kHC + h * kDim + lane];
#pragma unroll
      for (int off = 16; off > 0; off >>= 1) p += __shfl_xor(p, off, 32);
      const float dot = p * 0.17677669529663687f;  // 1/sqrt(32), wave-uniform
      if (dot > m) {                               // uniform branch
        const float c = expf(m - dot);             // 0 when m == -inf
        z *= c; o *= c; m = dot;
      }
      const float w = expf(dot - m);
      z += w;
      o += w * v[(size_t)src * kHC + h * kDim + lane];
    }
    acc += (z > 0.0f) ? (o / z) : 0.0f;            // empty segment -> 0 (matches JAX)
  }
  float val = acc * (1.0f / 3.0f) + s[(size_t)n * kDim + lane];
  val = fmaxf(val, 0.0f);
  x[(size_t)n * kDim + lane] = val;
  if (emit_mid) {  // softmax over dims 0..4 of the post-ReLU features
    const float x0 = __shfl(val, 0, 32), x1 = __shfl(val, 1, 32),
                x2 = __shfl(val, 2, 32), x3 = __shfl(val, 3, 32),
                x4 = __shfl(val, 4, 32);
    const float mx = fmaxf(fmaxf(fmaxf(x0, x1), fmaxf(x2, x3)), x4);
    const float e0 = expf(x0 - mx), e1 = expf(x1 - mx), e2 = expf(x2 - mx),
                e3 = expf(x3 - mx), e4 = expf(x4 - mx);
    const float inv = 1.0f / (e0 + e1 + e2 + e3 + e4);
    if (lane == 0) {
      float* mp = mid + (size_t)n * 5;
      mp[0] = e0 * inv; mp[1] = e1 * inv; mp[2] = e2 * inv;
      mp[3] = e3 * inv; mp[4] = e4 * inv;
    }
  }
}

// ---------- final head, softmax, atomic pool by graph id ----------
__global__ void head_pool_kernel(const float* __restrict__ x,
                                 const float* __restrict__ Wm,
                                 const float* __restrict__ bm,
                                 const int* __restrict__ batch,
                                 float* __restrict__ pooled,
                                 float* __restrict__ cnt) {
  const int n = (int)(blockIdx.x * blockDim.x + threadIdx.x);
  if (n >= kN) return;
  float y0 = bm[0], y1 = bm[1];
#pragma unroll
  for (int d = 0; d < kDim; ++d) {
    const float xv = x[(size_t)n * kDim + d];
    y0 += xv * Wm[d * 2 + 0];
    y1 += xv * Wm[d * 2 + 1];
  }
  y0 = fmaxf(y0, 0.0f);
  y1 = fmaxf(y1, 0.0f);
  const float mx = fmaxf(y0, y1);
  const float e0 = expf(y0 - mx), e1 = expf(y1 - mx);
  const float inv = 1.0f / (e0 + e1);
  const int g = batch[n];
  atomicAdd(&pooled[g * 2 + 0], e0 * inv);
  atomicAdd(&pooled[g * 2 + 1], e1 * inv);
  atomicAdd(&cnt[g], 1.0f);
}

__global__ void pool_finalize_kernel(const float* __restrict__ pooled,
                                     const float* __restrict__ cnt,
                                     float* __restrict__ outp) {
  const int g = (int)threadIdx.x;
  if (g < kG) {
    const float c = fmaxf(cnt[g], 1.0f);
    outp[g * 2 + 0] = pooled[g * 2 + 0] / c;
    outp[g * 2 + 1] = pooled[g * 2 + 1] / c;
  }
}

extern "C" void kernel_launch(void* const* d_in, const int* in_sizes, int n_in,
                              void* d_out, int out_size, void* d_ws, size_t ws_size,
                              hipStream_t stream) {
  (void)in_sizes; (void)n_in; (void)out_size; (void)ws_size;
  const float* x0 = (const float*)d_in[0];
  const float* Wq = (const float*)d_in[1];
  const float* bq = (const float*)d_in[2];
  const float* Wk = (const float*)d_in[3];
  const float* bk = (const float*)d_in[4];
  const float* Wv = (const float*)d_in[5];
  const float* bv = (const float*)d_in[6];
  const float* Ws = (const float*)d_in[7];
  const float* bs = (const float*)d_in[8];
  const float* Wm = (const float*)d_in[9];
  const float* bm = (const float*)d_in[10];
  const int*   ei = (const int*)d_in[11];
  const int*   batch = (const int*)d_in[12];
  float* outp = (float*)d_out;

  // workspace layout (4-byte elems, all segment sizes multiples of 16)
  float* ws = (float*)d_ws;
  float* xcur = ws;             ws += (size_t)kN * kDim;   // 640k
  float* q    = ws;             ws += (size_t)kN * kHC;    // 1.92M
  float* k    = ws;             ws += (size_t)kN * kHC;
  float* v    = ws;             ws += (size_t)kN * kHC;
  float* sbuf = ws;             ws += (size_t)kN * kDim;
  int* deg    = (int*)ws;       ws += kN;
  int* rowptr = (int*)ws;       ws += kN + 16;             // kN+1, padded
  int* cursor = (int*)ws;       ws += kN;
  int* esrc   = (int*)ws;       ws += kE;
  float* pooled = ws;           ws += 2 * kG;
  float* cnt  = ws;             ws += kG;

  const int projBlocks = (kTiles + 3) / 4;       // 4 waves of 32 per block
  const int edgeBlocks = (kE + 255) / 256;
  const int nodeBlocks = (kN + 255) / 256;
  const int attnBlocks = (kN * 32 + 255) / 256;  // one wave per node

  // ---- CSR of incoming edges (dst fixed for all layers): build once ----
  hipMemsetAsync(deg, 0, (size_t)kN * sizeof(int), stream);
  deg_kernel<<<edgeBlocks, 256, 0, stream>>>(ei, deg);
  scan_kernel<<<1, 1024, 0, stream>>>(deg, rowptr);
  hipMemcpyAsync(cursor, rowptr, (size_t)kN * sizeof(int),
                 hipMemcpyDeviceToDevice, stream);
  fill_kernel<<<edgeBlocks, 256, 0, stream>>>(ei, cursor, esrc);

  hipMemcpyAsync(xcur, x0, (size_t)kN * kDim * sizeof(float),
                 hipMemcpyDeviceToDevice, stream);

  for (int i = 0; i < 4; ++i) {
    proj_kernel<<<projBlocks, 128, 0, stream>>>(
        xcur,
        Wq + (size_t)i * kDim * kHC, bq + (size_t)i * kHC,
        Wk + (size_t)i * kDim * kHC, bk + (size_t)i * kHC,
        Wv + (size_t)i * kDim * kHC, bv + (size_t)i * kHC,
        Ws + (size_t)i * kDim * kDim, bs + (size_t)i * kDim,
        q, k, v, sbuf);
    attn_kernel<<<attnBlocks, 256, 0, stream>>>(
        rowptr, esrc, q, k, v, sbuf, xcur, outp, (i == 2) ? 1 : 0);
  }

  hipMemsetAsync(pooled, 0, 2 * kG * sizeof(float), stream);
  hipMemsetAsync(cnt, 0, kG * sizeof(float), stream);
  head_pool_kernel<<<nodeBlocks, 256, 0, stream>>>(xcur, Wm, bm, batch, pooled, cnt);
  pool_finalize_kernel<<<1, 32, 0, stream>>>(pooled, cnt, outp + (size_t)kN * 5);
}